// PointTransformerLayer_63178968924301
// MI455X (gfx1250) — compile-verified
//
#include <hip/hip_runtime.h>
#include <math.h>

// Problem constants (match reference)
#define N_PTS 32768
#define KNN   16
#define CH    64
#define CS    8          // CH / share_planes
#define EPSV  1e-5f

typedef float v2f __attribute__((ext_vector_type(2)));
typedef float v8f __attribute__((ext_vector_type(8)));

// ---- stats scratch layout (float offsets inside workspace) ----
#define ST_P_SUM   0     // [3]
#define ST_P_SQ    3     // [3]
#define ST_V_SUM   6     // [1]
#define ST_V_SQ    7     // [1]
#define ST_W1_SUM  8     // [64]
#define ST_W1_SQ   72    // [64]
#define ST_W2_SUM  136   // [8]
#define ST_W2_SQ   144   // [8]
#define ST_R_SUM   152   // [64]
#define ST_R_SQ    216   // [64]
#define ST_P_MEAN  280
#define ST_P_INV   283
#define ST_V_MEAN  286
#define ST_V_INV   287
#define ST_W1_MEAN 288
#define ST_W1_INV  352
#define ST_W2_MEAN 416
#define ST_W2_INV  424
#define ST_R_MEAN  432
#define ST_R_INV   496
#define ST_TOTAL   1024  // rounded

// --------------------------------------------------------------------------
// zero the stats accumulator region (must be deterministic per launch)
__global__ void k_zero(float* __restrict__ st) {
  int i = blockIdx.x * blockDim.x + threadIdx.x;
  if (i < ST_TOTAL) st[i] = 0.f;
}

// finalize BN stats: mean + rsqrt(var+eps) per channel
__global__ void k_finalize(float* __restrict__ st, int sum_off, int sq_off,
                           int mean_off, int inv_off, int nch, float cnt) {
  int c = threadIdx.x;
  if (c < nch) {
    float m = st[sum_off + c] / cnt;
    float var = st[sq_off + c] / cnt - m * m;
    st[mean_off + c] = m;
    st[inv_off + c] = rsqrtf(var + EPSV);
  }
}

// --------------------------------------------------------------------------
// QKV projection: [N,64] x [64,64] (+bias) for q,k,v.  One wave per 16-row
// tile (4 waves/block); V_WMMA_F32_16X16X4_F32, K in steps of 4, fully
// unrolled (64 WMMAs/wave, full fp32 precision).
__global__ __launch_bounds__(128)
void k_qkv_wmma(const float* __restrict__ feat,
                const float* __restrict__ wq, const float* __restrict__ bq,
                const float* __restrict__ wk, const float* __restrict__ bk,
                const float* __restrict__ wv, const float* __restrict__ bv,
                float* __restrict__ xq, float* __restrict__ xk,
                float* __restrict__ xv) {
  const int lane = threadIdx.x;                      // 0..31 (wave32)
  const int row0 = (blockIdx.x * 4 + threadIdx.y) * 16;
  const float* W; const float* B; float* O;
  if (blockIdx.y == 0)      { W = wq; B = bq; O = xq; }
  else if (blockIdx.y == 1) { W = wk; B = bk; O = xk; }
  else                      { W = wv; B = bv; O = xv; }

  const int r  = lane & 15;
  const int hi = lane >> 4;

  v8f acc[4] = {};
#pragma unroll
  for (int k0 = 0; k0 < CH; k0 += 4) {
    const int kk = k0 + 2 * hi;
    // A fragment (16x4 f32): lane<16 holds K=k0..k0+1, lane>=16 K=k0+2..k0+3
    v2f a;
    const float* ap = feat + (size_t)(row0 + r) * CH + kk;
    a.x = ap[0]; a.y = ap[1];
#pragma unroll
    for (int t = 0; t < 4; ++t) {
      v2f b;
      b.x = W[(size_t)kk * CH + t * 16 + r];
      b.y = W[(size_t)(kk + 1) * CH + t * 16 + r];
      acc[t] = __builtin_amdgcn_wmma_f32_16x16x4_f32(
          false, a, false, b, (short)0, acc[t], false, false);
    }
  }
#pragma unroll
  for (int t = 0; t < 4; ++t) {
    float bias = B[t * 16 + r];
#pragma unroll
    for (int rr = 0; rr < 8; ++rr)
      O[(size_t)(row0 + rr + 8 * hi) * CH + t * 16 + r] = acc[t][rr] + bias;
  }
}

// --------------------------------------------------------------------------
// p_lin[n,k,3] = (xyz[idx]-xyz) @ p1_w + p1_b ; v_lin[n,k] = vel[idx_v]*v1_w+v1_b
// plus BN accumulation for the 3 p-channels and 1 v-channel.
__global__ __launch_bounds__(256)
void k_pv_lin(const float* __restrict__ xyz, const float* __restrict__ vel,
              const int* __restrict__ idx, const int* __restrict__ idx_v,
              const float* __restrict__ p1_w, const float* __restrict__ p1_b,
              const float* __restrict__ v1_w, const float* __restrict__ v1_b,
              float* __restrict__ p_lin, float* __restrict__ v_lin,
              float* __restrict__ st) {
  __shared__ float s_acc[8];
  const int tid = threadIdx.x;
  if (tid < 8) s_acc[tid] = 0.f;
  __syncthreads();

  const int e = blockIdx.x * 256 + tid;   // N*K divisible by 256
  const int n = e >> 4;                   // e / KNN
  const int j = idx[e];
  float pr0 = xyz[(size_t)j * 3 + 0] - xyz[(size_t)n * 3 + 0];
  float pr1 = xyz[(size_t)j * 3 + 1] - xyz[(size_t)n * 3 + 1];
  float pr2 = xyz[(size_t)j * 3 + 2] - xyz[(size_t)n * 3 + 2];
  float pl[3];
#pragma unroll
  for (int c = 0; c < 3; ++c) {
    pl[c] = pr0 * p1_w[0 * 3 + c] + pr1 * p1_w[1 * 3 + c] +
            pr2 * p1_w[2 * 3 + c] + p1_b[c];
    p_lin[(size_t)e * 3 + c] = pl[c];
  }
  const int jv = idx_v[e];
  float vl = vel[jv] * v1_w[0] + v1_b[0];
  v_lin[e] = vl;

#pragma unroll
  for (int c = 0; c < 3; ++c) {
    atomicAdd(&s_acc[c], pl[c]);
    atomicAdd(&s_acc[3 + c], pl[c] * pl[c]);
  }
  atomicAdd(&s_acc[6], vl);
  atomicAdd(&s_acc[7], vl * vl);
  __syncthreads();
  if (tid < 8) atomicAdd(&st[ST_P_SUM + tid], s_acc[tid]);  // layout matches
}

// --------------------------------------------------------------------------
// enc(e,c) = p_enc + v_enc, recomputed from the tiny p_lin/v_lin tensors
// (4 FMAs/element; cheaper than storing a 134 MB tensor, L2-friendly).
__device__ __forceinline__ float enc_at(
    const float* __restrict__ p_lin, const float* __restrict__ v_lin,
    const float* __restrict__ st,
    const float* __restrict__ p2_w, const float* __restrict__ p2_b,
    const float* __restrict__ v2_w, const float* __restrict__ v2_b,
    const float* __restrict__ p_g, const float* __restrict__ p_bb,
    const float* __restrict__ v_g, const float* __restrict__ v_bb,
    int e, int c) {
  float acc = p2_b[c] + v2_b[c];
#pragma unroll
  for (int jj = 0; jj < 3; ++jj) {
    float x = (p_lin[(size_t)e * 3 + jj] - st[ST_P_MEAN + jj]) *
                  st[ST_P_INV + jj] * p_g[jj] + p_bb[jj];
    acc += fmaxf(x, 0.f) * p2_w[jj * CH + c];
  }
  float v = (v_lin[e] - st[ST_V_MEAN]) * st[ST_V_INV] * v_g[0] + v_bb[0];
  acc += fmaxf(v, 0.f) * v2_w[c];
  return acc;
}

// --------------------------------------------------------------------------
// Pass 1 of w-path BN: stats of  w = x_k[idx] - x_q + enc  over N*K per channel.
__global__ __launch_bounds__(256)
void k_w1_stats(const float* __restrict__ xq, const float* __restrict__ xk,
                const int* __restrict__ idx,
                const float* __restrict__ p_lin, const float* __restrict__ v_lin,
                const float* __restrict__ p2_w, const float* __restrict__ p2_b,
                const float* __restrict__ v2_w, const float* __restrict__ v2_b,
                const float* __restrict__ p_g, const float* __restrict__ p_bb,
                const float* __restrict__ v_g, const float* __restrict__ v_bb,
                float* __restrict__ st) {
  __shared__ float sred[256], sred2[256];
  const int c = threadIdx.x, y = threadIdx.y;
  float lsum = 0.f, lsq = 0.f;
  const int stride = gridDim.x * 4;
  for (int e0 = blockIdx.x * 4; e0 < N_PTS * KNN; e0 += stride) {
    const int e = e0 + y;
    const int n = e >> 4;
    const int j = idx[e];
    float w = xk[(size_t)j * CH + c] - xq[(size_t)n * CH + c] +
              enc_at(p_lin, v_lin, st, p2_w, p2_b, v2_w, v2_b,
                     p_g, p_bb, v_g, v_bb, e, c);
    lsum += w; lsq += w * w;
  }
  sred[y * 64 + c] = lsum;
  sred2[y * 64 + c] = lsq;
  __syncthreads();
  if (y == 0) {
    float s = sred[c] + sred[64 + c] + sred[128 + c] + sred[192 + c];
    float q = sred2[c] + sred2[64 + c] + sred2[128 + c] + sred2[192 + c];
    atomicAdd(&st[ST_W1_SUM + c], s);
    atomicAdd(&st[ST_W1_SQ + c], q);
  }
}

// --------------------------------------------------------------------------
// Pass 2: h[e,0..7] = relu(bn1(w)) @ w1_w + w1_b ; accumulate bn2 stats.
__global__ __launch_bounds__(256)
void k_h_stats(const float* __restrict__ xq, const float* __restrict__ xk,
               const int* __restrict__ idx,
               const float* __restrict__ p_lin, const float* __restrict__ v_lin,
               const float* __restrict__ p2_w, const float* __restrict__ p2_b,
               const float* __restrict__ v2_w, const float* __restrict__ v2_b,
               const float* __restrict__ p_g, const float* __restrict__ p_bb,
               const float* __restrict__ v_g, const float* __restrict__ v_bb,
               const float* __restrict__ w1_g, const float* __restrict__ w1_bb,
               const float* __restrict__ w1_w, const float* __restrict__ w1_bias,
               float* __restrict__ h, float* __restrict__ st) {
  __shared__ float s_w[4][64];
  __shared__ float s_p[4][CS][2];
  const int c = threadIdx.x, y = threadIdx.y;
  float hsum = 0.f, hsq = 0.f;
  const int stride = gridDim.x * 4;
  for (int e0 = blockIdx.x * 4; e0 < N_PTS * KNN; e0 += stride) {
    const int e = e0 + y;
    const int n = e >> 4;
    const int j = idx[e];
    float w = xk[(size_t)j * CH + c] - xq[(size_t)n * CH + c] +
              enc_at(p_lin, v_lin, st, p2_w, p2_b, v2_w, v2_b,
                     p_g, p_bb, v_g, v_bb, e, c);
    w = fmaxf((w - st[ST_W1_MEAN + c]) * st[ST_W1_INV + c] * w1_g[c] + w1_bb[c],
              0.f);
    s_w[y][c] = w;
    __syncthreads();
    if (c < CS) {
      float acc = w1_bias[c];
#pragma unroll
      for (int q = 0; q < CH; ++q) acc += s_w[y][q] * w1_w[q * CS + c];
      h[(size_t)e * CS + c] = acc;
      hsum += acc; hsq += acc * acc;
    }
    __syncthreads();
  }
  if (c < CS) { s_p[y][c][0] = hsum; s_p[y][c][1] = hsq; }
  __syncthreads();
  if (y == 0 && c < CS) {
    float s = s_p[0][c][0] + s_p[1][c][0] + s_p[2][c][0] + s_p[3][c][0];
    float q = s_p[0][c][1] + s_p[1][c][1] + s_p[2][c][1] + s_p[3][c][1];
    atomicAdd(&st[ST_W2_SUM + c], s);
    atomicAdd(&st[ST_W2_SQ + c], q);
  }
}

// --------------------------------------------------------------------------
// logits = relu(bn2(h)) @ w2_w + w2_b ; softmax over K per plane;
// out = sum_k (x_v[idx]+enc) * soft[k, c%8] + features ; rho BN stats.
__global__ __launch_bounds__(64)
void k_attn(const float* __restrict__ xv, const float* __restrict__ feat,
            const int* __restrict__ idx,
            const float* __restrict__ p_lin, const float* __restrict__ v_lin,
            const float* __restrict__ p2_w, const float* __restrict__ p2_b,
            const float* __restrict__ v2_w, const float* __restrict__ v2_b,
            const float* __restrict__ p_g, const float* __restrict__ p_bb,
            const float* __restrict__ v_g, const float* __restrict__ v_bb,
            const float* __restrict__ w2_g, const float* __restrict__ w2_bb,
            const float* __restrict__ w2_w, const float* __restrict__ w2_bias,
            const float* __restrict__ h, float* __restrict__ pre_rho,
            float* __restrict__ st) {
  __shared__ float s_soft[KNN * CS];
  const int t = threadIdx.x;
  float rsum = 0.f, rsq = 0.f;
  for (int n = blockIdx.x; n < N_PTS; n += gridDim.x) {
    if (t < KNN) {
      const int k = t;
      float hb[CS];
#pragma unroll
      for (int q = 0; q < CS; ++q) {
        float x = h[((size_t)n * KNN + k) * CS + q];
        hb[q] = fmaxf((x - st[ST_W2_MEAN + q]) * st[ST_W2_INV + q] * w2_g[q] +
                          w2_bb[q], 0.f);
      }
#pragma unroll
      for (int p = 0; p < CS; ++p) {
        float l = w2_bias[p];
#pragma unroll
        for (int q = 0; q < CS; ++q) l += hb[q] * w2_w[q * CS + p];
        s_soft[k * CS + p] = l;
      }
    }
    __syncthreads();
    if (t < CS) {
      const int p = t;
      float mx = -3.0e38f;
      for (int k = 0; k < KNN; ++k) mx = fmaxf(mx, s_soft[k * CS + p]);
      float ex[KNN], sum = 0.f;
      for (int k = 0; k < KNN; ++k) { ex[k] = expf(s_soft[k * CS + p] - mx); sum += ex[k]; }
      float inv = 1.f / sum;
      for (int k = 0; k < KNN; ++k) s_soft[k * CS + p] = ex[k] * inv;
    }
    __syncthreads();
    {
      const int c = t;
      float acc = 0.f;
      for (int k = 0; k < KNN; ++k) {
        const int e = n * KNN + k;
        const int j = idx[e];
        acc += (xv[(size_t)j * CH + c] +
                enc_at(p_lin, v_lin, st, p2_w, p2_b, v2_w, v2_b,
                       p_g, p_bb, v_g, v_bb, e, c)) *
               s_soft[k * CS + (c & 7)];
      }
      float val = acc + feat[(size_t)n * CH + c];
      pre_rho[(size_t)n * CH + c] = val;
      rsum += val; rsq += val * val;
    }
    __syncthreads();
  }
  atomicAdd(&st[ST_R_SUM + t], rsum);
  atomicAdd(&st[ST_R_SQ + t], rsq);
}

// --------------------------------------------------------------------------
// Final: out = relu(bn(pre_rho)) @ rho_w + rho_b  via WMMA (BN+ReLU fused
// into the A-fragment load).  4 waves/block, one 16-row tile per wave.
__global__ __launch_bounds__(128)
void k_rho_wmma(const float* __restrict__ pre_rho, const float* __restrict__ st,
                const float* __restrict__ rho_g, const float* __restrict__ rho_bbn,
                const float* __restrict__ rho_w, const float* __restrict__ rho_bias,
                float* __restrict__ out) {
  const int lane = threadIdx.x;
  const int row0 = (blockIdx.x * 4 + threadIdx.y) * 16;
  const int r = lane & 15, hi = lane >> 4;
  v8f acc[4] = {};
#pragma unroll
  for (int k0 = 0; k0 < CH; k0 += 4) {
    const int kk = k0 + 2 * hi;
    float x0 = pre_rho[(size_t)(row0 + r) * CH + kk];
    float x1 = pre_rho[(size_t)(row0 + r) * CH + kk + 1];
    v2f a;
    a.x = fmaxf((x0 - st[ST_R_MEAN + kk]) * st[ST_R_INV + kk] * rho_g[kk] +
                    rho_bbn[kk], 0.f);
    a.y = fmaxf((x1 - st[ST_R_MEAN + kk + 1]) * st[ST_R_INV + kk + 1] *
                        rho_g[kk + 1] + rho_bbn[kk + 1], 0.f);
#pragma unroll
    for (int tt = 0; tt < 4; ++tt) {
      v2f b;
      b.x = rho_w[(size_t)kk * CH + tt * 16 + r];
      b.y = rho_w[(size_t)(kk + 1) * CH + tt * 16 + r];
      acc[tt] = __builtin_amdgcn_wmma_f32_16x16x4_f32(
          false, a, false, b, (short)0, acc[tt], false, false);
    }
  }
#pragma unroll
  for (int tt = 0; tt < 4; ++tt) {
    float bias = rho_bias[tt * 16 + r];
#pragma unroll
    for (int rr = 0; rr < 8; ++rr)
      out[(size_t)(row0 + rr + 8 * hi) * CH + tt * 16 + r] = acc[tt][rr] + bias;
  }
}

// --------------------------------------------------------------------------
extern "C" void kernel_launch(void* const* d_in, const int* in_sizes, int n_in,
                              void* d_out, int out_size, void* d_ws,
                              size_t ws_size, hipStream_t stream) {
  (void)in_sizes; (void)n_in; (void)out_size; (void)ws_size;
  const float* xyz    = (const float*)d_in[0];
  const float* feat   = (const float*)d_in[1];
  const float* vel    = (const float*)d_in[2];
  const float* wq     = (const float*)d_in[3];
  const float* bq     = (const float*)d_in[4];
  const float* wk     = (const float*)d_in[5];
  const float* bk     = (const float*)d_in[6];
  const float* wv     = (const float*)d_in[7];
  const float* bv     = (const float*)d_in[8];
  const float* p1_w   = (const float*)d_in[9];
  const float* p1_b   = (const float*)d_in[10];
  const float* p_bn_g = (const float*)d_in[11];
  const float* p_bn_b = (const float*)d_in[12];
  const float* p2_w   = (const float*)d_in[13];
  const float* p2_b   = (const float*)d_in[14];
  const float* v1_w   = (const float*)d_in[15];
  const float* v1_b   = (const float*)d_in[16];
  const float* v_bn_g = (const float*)d_in[17];
  const float* v_bn_b = (const float*)d_in[18];
  const float* v2_w   = (const float*)d_in[19];
  const float* v2_b   = (const float*)d_in[20];
  const float* w_bn1_g = (const float*)d_in[21];
  const float* w_bn1_b = (const float*)d_in[22];
  const float* w1_w   = (const float*)d_in[23];
  const float* w1_b   = (const float*)d_in[24];
  const float* w_bn2_g = (const float*)d_in[25];
  const float* w_bn2_b = (const float*)d_in[26];
  const float* w2_w   = (const float*)d_in[27];
  const float* w2_b   = (const float*)d_in[28];
  const float* rho_bn_g = (const float*)d_in[29];
  const float* rho_bn_b = (const float*)d_in[30];
  const float* rho_w  = (const float*)d_in[31];
  const float* rho_b  = (const float*)d_in[32];
  const int* idx      = (const int*)d_in[33];
  const int* idx_v    = (const int*)d_in[34];
  float* out = (float*)d_out;

  // workspace carve-up (~59 MB; fits comfortably in MI455X's 192 MB L2)
  float* ws      = (float*)d_ws;
  float* st      = ws;
  float* xq      = ws + ST_TOTAL;
  float* xk      = xq + (size_t)N_PTS * CH;
  float* xv      = xk + (size_t)N_PTS * CH;
  float* p_lin   = xv + (size_t)N_PTS * CH;
  float* v_lin   = p_lin + (size_t)N_PTS * KNN * 3;
  float* hbuf    = v_lin + (size_t)N_PTS * KNN;
  float* pre_rho = hbuf + (size_t)N_PTS * KNN * CS;

  const float cntNK = (float)(N_PTS * KNN);
  const float cntN  = (float)N_PTS;

  k_zero<<<1, 1024, 0, stream>>>(st);
  k_qkv_wmma<<<dim3(N_PTS / 64, 3), dim3(32, 4), 0, stream>>>(
      feat, wq, bq, wk, bk, wv, bv, xq, xk, xv);
  k_pv_lin<<<N_PTS * KNN / 256, 256, 0, stream>>>(
      xyz, vel, idx, idx_v, p1_w, p1_b, v1_w, v1_b, p_lin, v_lin, st);
  k_finalize<<<1, 32, 0, stream>>>(st, ST_P_SUM, ST_P_SQ, ST_P_MEAN, ST_P_INV, 3, cntNK);
  k_finalize<<<1, 32, 0, stream>>>(st, ST_V_SUM, ST_V_SQ, ST_V_MEAN, ST_V_INV, 1, cntNK);
  k_w1_stats<<<2048, dim3(64, 4), 0, stream>>>(
      xq, xk, idx, p_lin, v_lin, p2_w, p2_b, v2_w, v2_b,
      p_bn_g, p_bn_b, v_bn_g, v_bn_b, st);
  k_finalize<<<1, 64, 0, stream>>>(st, ST_W1_SUM, ST_W1_SQ, ST_W1_MEAN, ST_W1_INV, 64, cntNK);
  k_h_stats<<<2048, dim3(64, 4), 0, stream>>>(
      xq, xk, idx, p_lin, v_lin, p2_w, p2_b, v2_w, v2_b,
      p_bn_g, p_bn_b, v_bn_g, v_bn_b, w_bn1_g, w_bn1_b, w1_w, w1_b, hbuf, st);
  k_finalize<<<1, 32, 0, stream>>>(st, ST_W2_SUM, ST_W2_SQ, ST_W2_MEAN, ST_W2_INV, CS, cntNK);
  k_attn<<<2048, 64, 0, stream>>>(
      xv, feat, idx, p_lin, v_lin, p2_w, p2_b, v2_w, v2_b,
      p_bn_g, p_bn_b, v_bn_g, v_bn_b, w_bn2_g, w_bn2_b, w2_w, w2_b,
      hbuf, pre_rho, st);
  k_finalize<<<1, 64, 0, stream>>>(st, ST_R_SUM, ST_R_SQ, ST_R_MEAN, ST_R_INV, 64, cntN);
  k_rho_wmma<<<N_PTS / 64, dim3(32, 4), 0, stream>>>(
      pre_rho, st, rho_bn_g, rho_bn_b, rho_w, rho_b, out);
}